// Sampler_88364657148060
// MI455X (gfx1250) — compile-verified
//
#include <hip/hip_runtime.h>

// NeRF-style volume compositing for MI455X (gfx1250, wave32).
//
// Strategy: one wave32 per ray. N=128 samples processed in 4 chunks of 32.
//   1 - alpha = exp(-density*delta)  = "om"
//   P = inclusive prefix product of om across the 32 lanes (shuffle scan)
//   E = exclusive prefix product ( = shfl_up(P,1), lane0 -> 1 )
//   weight = T_running * (E - P)      [ == alpha * exclusive transmittance ]
//   T_running *= broadcast(P, lane 31) per chunk.
// All big streams use non-temporal loads (read-once, don't churn the 192MB L2).
// Memory-bandwidth bound: 288 MB @ 23.3 TB/s => ~12.5us floor; WMMA is
// inapplicable (batched per-ray contraction, not a GEMM).

#define NS  128         // samples per ray (fixed by reference)
#define WPB 8           // waves (rays) per block
#define TPB (WPB * 32)  // 256 threads = 8 wave32

typedef float v4f __attribute__((ext_vector_type(4)));

__device__ __forceinline__ float wave_sum(float v) {
#pragma unroll
    for (int off = 16; off > 0; off >>= 1)
        v += __shfl_xor(v, off, 32);
    return v;
}

__global__ __launch_bounds__(TPB) void nerf_composite_kernel(
    const float* __restrict__ density,
    const float* __restrict__ deltas,
    const float* __restrict__ zs,
    const float* __restrict__ rgbs,
    const float* __restrict__ normals,
    float* __restrict__ out,
    int B)
{
    const int lane = threadIdx.x & 31;
    const int wv   = threadIdx.x >> 5;
    const int b    = blockIdx.x * WPB + wv;
    if (b >= B) return;

    const float* dns = density + (size_t)b * NS;
    const float* dlt = deltas  + (size_t)b * NS;
    const float* zz  = zs      + (size_t)b * NS;
    const float* rgb = rgbs    + (size_t)b * NS * 3;
    const float* nrm = normals + (size_t)b * NS * 3;

    float T = 1.0f;                                   // running transmittance
    float sr = 0.f, sg = 0.f, sb = 0.f, sz = 0.f;
    float snx = 0.f, sny = 0.f, snz = 0.f, sa = 0.f;

#pragma unroll
    for (int c = 0; c < NS / 32; ++c) {
        const int n = c * 32 + lane;

        // Prefetch the fat per-sample feature streams one chunk ahead
        // (gfx1250 global_prefetch_b8; speculative streaming hint).
        if (c + 1 < NS / 32) {
            __builtin_prefetch(rgb + 3 * (n + 32), 0, 0);
            __builtin_prefetch(nrm + 3 * (n + 32), 0, 0);
        }

        // Read-once streams: non-temporal so they bypass-priority the caches.
        float d0 = __builtin_nontemporal_load(dns + n);
        float d1 = __builtin_nontemporal_load(dlt + n);
        float z  = __builtin_nontemporal_load(zz  + n);
        float r0 = __builtin_nontemporal_load(rgb + 3 * n + 0);
        float r1 = __builtin_nontemporal_load(rgb + 3 * n + 1);
        float r2 = __builtin_nontemporal_load(rgb + 3 * n + 2);
        float m0 = __builtin_nontemporal_load(nrm + 3 * n + 0);
        float m1 = __builtin_nontemporal_load(nrm + 3 * n + 1);
        float m2 = __builtin_nontemporal_load(nrm + 3 * n + 2);

        // 1 - alpha = exp(-sigma*delta); v_exp_f32 path.
        float om = __expf(-(d0 * d1));

        // Inclusive prefix product across the wave32 (Hillis-Steele, 5 steps).
        float P = om;
#pragma unroll
        for (int off = 1; off < 32; off <<= 1) {
            float t = __shfl_up(P, off, 32);
            P *= (lane >= off) ? t : 1.0f;
        }
        // Exclusive prefix product.
        float E = __shfl_up(P, 1, 32);
        E = (lane == 0) ? 1.0f : E;

        // weight = alpha * exclusive_transmittance = T * (E - E*om) = T*(E - P)
        float w = T * (E - P);

        sr  += w * r0;  sg  += w * r1;  sb  += w * r2;
        snx += w * m0;  sny += w * m1;  snz += w * m2;
        sz  += w * z;   sa  += w;

        // Carry transmittance across chunks.
        T *= __shfl(P, 31, 32);
    }

    // Cross-lane reductions (butterfly) for the 8 composites.
    sr  = wave_sum(sr);   sg  = wave_sum(sg);   sb  = wave_sum(sb);
    snx = wave_sum(snx);  sny = wave_sum(sny);  snz = wave_sum(snz);
    sz  = wave_sum(sz);   sa  = wave_sum(sa);

    if (lane == 0) {
        float depth = sz / (sa + 1e-8f);
        v4f o0 = { sr,  sg,  sb,  depth };
        v4f o1 = { snx, sny, snz, sa    };
        v4f* o = (v4f*)(out + (size_t)b * 8);
        __builtin_nontemporal_store(o0, o);
        __builtin_nontemporal_store(o1, o + 1);
    }
}

extern "C" void kernel_launch(void* const* d_in, const int* in_sizes, int n_in,
                              void* d_out, int out_size, void* d_ws, size_t ws_size,
                              hipStream_t stream) {
    // Inputs (setup_inputs order): density, deltas, zs, raw_rgbs, raw_normals
    const float* density = (const float*)d_in[0];
    const float* deltas  = (const float*)d_in[1];
    const float* zs      = (const float*)d_in[2];
    const float* rgbs    = (const float*)d_in[3];
    const float* normals = (const float*)d_in[4];
    float* out = (float*)d_out;

    const int B = in_sizes[0] / NS;   // 65536 for the reference shapes
    dim3 grid((B + WPB - 1) / WPB);
    nerf_composite_kernel<<<grid, TPB, 0, stream>>>(density, deltas, zs, rgbs,
                                                    normals, out, B);
}